// GCNBlock_75230647157512
// MI455X (gfx1250) — compile-verified
//
#include <hip/hip_runtime.h>

typedef float v2f __attribute__((ext_vector_type(2)));
typedef float v8f __attribute__((ext_vector_type(8)));

__device__ __forceinline__ void hw_atomic_add_f32(float* p, float v) {
    // Lowers to non-returning global_atomic_add_f32 (agent scope, relaxed).
    __hip_atomic_fetch_add(p, v, __ATOMIC_RELAXED, __HIP_MEMORY_SCOPE_AGENT);
}

// ---------------------------------------------------------------------------
// Degree / normalization
// ---------------------------------------------------------------------------
__global__ void init_deg_kernel(float* __restrict__ deg, int n) {
    int i = blockIdx.x * blockDim.x + threadIdx.x;
    if (i < n) deg[i] = 1.0f;  // self-loop contributes 1 to every node
}

__global__ void deg_edges_kernel(const int* __restrict__ col,
                                 float* __restrict__ deg, int e) {
    int i = blockIdx.x * blockDim.x + threadIdx.x;
    if (i < e) hw_atomic_add_f32(&deg[col[i]], 1.0f);
}

__global__ void make_dinv_kernel(const float* __restrict__ deg,
                                 float* __restrict__ dinv, int n) {
    int i = blockIdx.x * blockDim.x + threadIdx.x;
    if (i < n) {
        float d = deg[i];
        dinv[i] = (d > 0.0f) ? rsqrtf(d) : 0.0f;
    }
}

// ---------------------------------------------------------------------------
// Dense transform h = x @ W[l] using V_WMMA_F32_16X16X4_F32 (wave32).
// Each wave computes a 16x32 output tile (two 16x16 accumulators sharing one
// A fragment -> halves A traffic vs one-tile-per-wave). A block of 8 waves
// covers 128 rows; blockIdx.y selects the 32-wide column tile. The two
// 128x16 W slices live in separate LDS arrays so each ds_load keeps the
// conflict-free 64B row stride.
// ---------------------------------------------------------------------------
__global__ __launch_bounds__(256) void gcn_gemm_kernel(
    const float* __restrict__ X, const float* __restrict__ Wl,
    float* __restrict__ H, int n) {
    __shared__ float Ws0[128 * 16];
    __shared__ float Ws1[128 * 16];

    const int tid = threadIdx.x;
    const int colBase = blockIdx.y * 32;

    // cooperative load of W[:, colBase : colBase+32] into two 128x16 slices
    for (int i = tid; i < 128 * 16; i += 256) {
        int r = i >> 4;
        int c = i & 15;
        Ws0[i] = Wl[r * 128 + colBase + c];
        Ws1[i] = Wl[r * 128 + colBase + 16 + c];
    }
    __syncthreads();

    const int wave = tid >> 5;      // wave32
    const int lane = tid & 31;
    const int tileM = blockIdx.x * 8 + wave;
    const int nTiles = n >> 4;      // N divisible by 16 (50000/16 = 3125)
    if (tileM >= nTiles) return;    // wave-uniform guard; EXEC stays all-1s

    const int ml    = lane & 15;          // M index within tile (A) / N (B,D)
    const int khalf = (lane >> 4) << 1;   // lanes 0-15 -> K 0/1, 16-31 -> K 2/3
    const int m = (tileM << 4) + ml;
    const float* __restrict__ xrow = X + (size_t)m * 128;

    v8f acc0 = {};
    v8f acc1 = {};
    #pragma unroll 4
    for (int k = 0; k < 128; k += 4) {
        const int k0 = k + khalf;
        v2f a, b0, b1;
        a.x  = xrow[k0];
        a.y  = xrow[k0 + 1];
        b0.x = Ws0[k0 * 16 + ml];
        b0.y = Ws0[(k0 + 1) * 16 + ml];
        b1.x = Ws1[k0 * 16 + ml];
        b1.y = Ws1[(k0 + 1) * 16 + ml];
        // 8 args: (neg_a, A, neg_b, B, c_mod, C, reuse_a, reuse_b)
        acc0 = __builtin_amdgcn_wmma_f32_16x16x4_f32(
            false, a, false, b0, (short)0, acc0, false, false);
        acc1 = __builtin_amdgcn_wmma_f32_16x16x4_f32(
            false, a, false, b1, (short)0, acc1, false, false);
    }

    // D layout: VGPR i -> row i (lanes 0-15) / row i+8 (lanes 16-31)
    const int rowBase = (tileM << 4) + ((lane >> 4) << 3);
    float* __restrict__ out0 = H + (size_t)rowBase * 128 + colBase + ml;
    #pragma unroll
    for (int i = 0; i < 8; ++i) {
        out0[(size_t)i * 128]      = acc0[i];
        out0[(size_t)i * 128 + 16] = acc1[i];
    }
}

// ---------------------------------------------------------------------------
// Zero an fp32 buffer (float4 stores)
// ---------------------------------------------------------------------------
__global__ void zero_f4_kernel(float4* __restrict__ p, long long n4) {
    long long i = (long long)blockIdx.x * blockDim.x + threadIdx.x;
    if (i < n4) p[i] = make_float4(0.f, 0.f, 0.f, 0.f);
}

// ---------------------------------------------------------------------------
// Scatter-add: one wave per (edge or self-loop); each lane moves a float4
// (32 lanes x 16 B = one full 512 B feature row, fully coalesced).
// agg[col] += h[row] * dinv[row]*dinv[col] via hardware f32 atomics (L2).
// ---------------------------------------------------------------------------
__global__ __launch_bounds__(256) void scatter_kernel(
    const float* __restrict__ H, const int* __restrict__ row,
    const int* __restrict__ col, const float* __restrict__ dinv,
    float* __restrict__ agg, int e, int n) {
    long long t = (long long)blockIdx.x * blockDim.x + threadIdx.x;
    long long w = t >> 5;               // wave-sized work item = one edge
    int lane = (int)(t & 31);
    long long total = (long long)e + (long long)n;
    if (w >= total) return;

    int r, c;
    if (w < (long long)e) {
        r = row[w];
        c = col[w];
    } else {
        r = c = (int)(w - (long long)e);  // self-loop
    }
    float s = dinv[r] * dinv[c];

    const float4 v = ((const float4*)(H + (size_t)r * 128))[lane];
    float* dst = agg + (size_t)c * 128 + lane * 4;
    hw_atomic_add_f32(dst + 0, v.x * s);
    hw_atomic_add_f32(dst + 1, v.y * s);
    hw_atomic_add_f32(dst + 2, v.z * s);
    hw_atomic_add_f32(dst + 3, v.w * s);
}

// ---------------------------------------------------------------------------
// x = relu(agg + b[l])  (in place)
// ---------------------------------------------------------------------------
__global__ void bias_relu_kernel(float* __restrict__ x,
                                 const float* __restrict__ b,
                                 long long nelem) {
    long long i = (long long)blockIdx.x * blockDim.x + threadIdx.x;
    if (i < nelem) {
        float v = x[i] + b[(int)(i & 127)];
        x[i] = (v > 0.0f) ? v : 0.0f;
    }
}

// ---------------------------------------------------------------------------
extern "C" void kernel_launch(void* const* d_in, const int* in_sizes, int n_in,
                              void* d_out, int out_size, void* d_ws, size_t ws_size,
                              hipStream_t stream) {
    const float* x    = (const float*)d_in[0];   // [N,128]
    const int*   edge = (const int*)d_in[1];     // [2,E] flat
    // d_in[2] batch_index, d_in[3] node_rankings: unused (NullControl = 0)
    const float* W    = (const float*)d_in[4];   // [4,128,128]
    const float* b    = (const float*)d_in[5];   // [4,128]

    const int n = in_sizes[0] / 128;
    const int e = in_sizes[1] / 2;
    const int* rowp = edge;       // sources
    const int* colp = edge + e;   // destinations

    // workspace: h[N*128] | ping[N*128] | deg[N] | dinv[N]
    float* h    = (float*)d_ws;
    float* ping = h + (size_t)n * 128;
    float* deg  = ping + (size_t)n * 128;
    float* dinv = deg + n;
    float* out  = (float*)d_out;

    const int B = 256;

    init_deg_kernel<<<(n + B - 1) / B, B, 0, stream>>>(deg, n);
    deg_edges_kernel<<<(e + B - 1) / B, B, 0, stream>>>(colp, deg, e);
    make_dinv_kernel<<<(n + B - 1) / B, B, 0, stream>>>(deg, dinv, n);

    const long long nelem = (long long)n * 128;
    const long long n4 = nelem / 4;
    const long long scatterThreads = ((long long)e + n) * 32;
    const int scatterBlocks = (int)((scatterThreads + B - 1) / B);
    dim3 ggrid((n / 16 + 7) / 8, 4);  // 32-wide column tiles

    const float* xin = x;
    for (int l = 0; l < 4; ++l) {
        gcn_gemm_kernel<<<ggrid, B, 0, stream>>>(xin, W + (size_t)l * 128 * 128, h, n);
        // agg(l) may reuse the buffer holding xin(l): gemm has already
        // consumed it (stream ordering). Final layer goes straight to d_out.
        float* agg = (l == 3) ? out : ping;
        zero_f4_kernel<<<(int)((n4 + B - 1) / B), B, 0, stream>>>((float4*)agg, n4);
        scatter_kernel<<<scatterBlocks, B, 0, stream>>>(h, rowp, colp, dinv, agg, e, n);
        bias_relu_kernel<<<(int)((nelem + B - 1) / B), B, 0, stream>>>(agg, b + (size_t)l * 128, nelem);
        xin = agg;
    }
}